// BLinear_39694087749724
// MI455X (gfx1250) — compile-verified
//
#include <hip/hip_runtime.h>
#include <math.h>

typedef __attribute__((ext_vector_type(2))) float v2f;
typedef __attribute__((ext_vector_type(8))) float v8f;

#define DIN 128
#define DOUT 128
#define WSTRIDE 132               // padded LDS row stride (floats): banks 4n -> conflict-free b64
#define MAXN 0.996f               // (1 - 4e-3)/sqrt(c), c = 1
#define MINN 1e-15f

// gfx1250 has a hardware TANH trans op; use it when the builtin exists.
#if __has_builtin(__builtin_amdgcn_tanhf)
__device__ __forceinline__ float fast_tanh(float x) { return __builtin_amdgcn_tanhf(x); }
#elif __has_builtin(__builtin_amdgcn_tanh_f32)
__device__ __forceinline__ float fast_tanh(float x) { return __builtin_amdgcn_tanh_f32(x); }
#else
__device__ __forceinline__ float fast_tanh(float x) { return tanhf(x); }
#endif

__device__ __forceinline__ float artanh_f(float x) {
    x = fminf(fmaxf(x, -1.0f + 1e-7f), 1.0f - 1e-7f);
    return 0.5f * (log1pf(x) - log1pf(-x));
}

// sum across each 16-lane half independently (rows 0..7 live in lanes 0-15, rows 8..15 in 16-31)
__device__ __forceinline__ float hsum16(float v) {
    v += __shfl_xor(v, 1, 32);
    v += __shfl_xor(v, 2, 32);
    v += __shfl_xor(v, 4, 32);
    v += __shfl_xor(v, 8, 32);
    return v;
}

__global__ __launch_bounds__(256) void hyp_linear_kernel(
    const float* __restrict__ X,   // [N, 128]
    const float* __restrict__ W,   // [128, 128] row-major [out][in]
    const float* __restrict__ B,   // [128]
    float* __restrict__ OUT,       // [N, 128]
    int nrows)
{
    extern __shared__ float smem[];
    float* ldsW  = smem;                    // 128 * 132
    float* ldsHB = smem + DOUT * WSTRIDE;   // 128
    float* ldsY2 = ldsHB + DOUT;            // 1

    const int tid  = threadIdx.x;
    const int lane = tid & 31;
    const int wave = tid >> 5;
    const int n15  = lane & 15;
    const int kp   = lane >> 4;             // which K-half this lane supplies

    // ---- stage W into padded LDS (float4 copies, 16B-aligned since 132*4=528 ≡ 0 mod 16) ----
    for (int i = tid; i < DOUT * (DIN / 4); i += 256) {
        const int r  = i >> 5;              // DIN/4 == 32
        const int c4 = (i & 31) << 2;
        const float4 wv = *(const float4*)(W + r * DIN + c4);
        *(float4*)(&ldsW[r * WSTRIDE + c4]) = wv;
    }

    // ---- hyperbolic bias: hb = proj(expmap0(b)), y2 = ||hb||^2 (wave 0) ----
    if (wave == 0) {
        float bl[4];
        float ss = 0.f;
        #pragma unroll
        for (int j = 0; j < 4; ++j) { bl[j] = B[lane + 32 * j]; ss = fmaf(bl[j], bl[j], ss); }
        #pragma unroll
        for (int m = 1; m < 32; m <<= 1) ss += __shfl_xor(ss, m, 32);
        const float bn = fmaxf(sqrtf(ss), MINN);
        const float t  = fast_tanh(bn);
        const float pf = (t > MAXN) ? (MAXN / fmaxf(t, MINN)) : 1.0f;
        const float sc = (t / bn) * pf;
        float ss2 = 0.f;
        #pragma unroll
        for (int j = 0; j < 4; ++j) {
            const float h = bl[j] * sc;
            ldsHB[lane + 32 * j] = h;
            ss2 = fmaf(h, h, ss2);
        }
        #pragma unroll
        for (int m = 1; m < 32; m <<= 1) ss2 += __shfl_xor(ss2, m, 32);
        if (lane == 0) ldsY2[0] = ss2;
    }
    __syncthreads();

    // ---- GEMM: mx = x @ W^T for a 16x128 strip, f32 WMMA 16x16x4, K-loop over 128 ----
    const long row0 = (long)blockIdx.x * 128 + (long)wave * 16;
    const float* xrow = X + (row0 + n15) * DIN;

    v8f acc[8];
    #pragma unroll
    for (int t = 0; t < 8; ++t) acc[t] = (v8f){0.f, 0.f, 0.f, 0.f, 0.f, 0.f, 0.f, 0.f};

    float xsq = 0.f;   // partial ||x_row||^2 (this lane's K-half)

    #pragma unroll 4
    for (int kb = 0; kb < DIN; kb += 4) {
        const int k = kb + 2 * kp;
        const float2 a2 = *(const float2*)(xrow + k);
        xsq = fmaf(a2.x, a2.x, xsq);
        xsq = fmaf(a2.y, a2.y, xsq);
        v2f A; A[0] = a2.x; A[1] = a2.y;

        // load all 8 B-fragments into distinct registers first so multiple
        // ds_load_2addr_b64 can be in flight (avoid wait-dscnt-0 serialization)
        v2f Bf[8];
        #pragma unroll
        for (int t = 0; t < 8; ++t) {
            const float2 b2 = *(const float2*)(&ldsW[(t * 16 + n15) * WSTRIDE + k]);
            Bf[t][0] = b2.x; Bf[t][1] = b2.y;
        }
        #pragma unroll
        for (int t = 0; t < 8; ++t) {
            acc[t] = __builtin_amdgcn_wmma_f32_16x16x4_f32(
                false, A, false, Bf[t], (short)0, acc[t], false, false);
        }
    }

    // combine the two K-halves of ||x_row||^2: lanes m and m+16 now hold row m's full sum
    xsq += __shfl_xor(xsq, 16, 32);
    const float y2 = ldsY2[0];

    // ---- mobius_matvec scale + proj: res = tanh(mn/xn * artanh(xn)) * mx / mn, clipped to ball ----
    float scl[8], rnrm[8];
    #pragma unroll
    for (int v = 0; v < 8; ++v) {
        float p = 0.f;
        #pragma unroll
        for (int t = 0; t < 8; ++t) p = fmaf(acc[t][v], acc[t][v], p);
        const float mn2 = hsum16(p);
        const int   m   = v + 8 * kp;
        const float xn2 = __shfl(xsq, m, 32);
        const float xn  = fmaxf(sqrtf(xn2), MINN);
        const float mn  = fmaxf(sqrtf(mn2), MINN);
        const float th  = fast_tanh((mn / xn) * artanh_f(xn));
        const float pf  = (th > MAXN) ? (MAXN / th) : 1.0f;
        const float tn  = th * pf;                       // ||res|| after proj
        scl[v]  = (mn2 == 0.f) ? 0.f : (tn / mn);        // cond: all(mx==0) -> 0
        rnrm[v] = (mn2 == 0.f) ? 0.f : tn;
    }

    // ---- mobius_add(res, hb) ----
    float hbf[8];
    #pragma unroll
    for (int t = 0; t < 8; ++t) hbf[t] = ldsHB[t * 16 + n15];

    float xyp[8];
    #pragma unroll
    for (int v = 0; v < 8; ++v) {
        float p = 0.f;
        #pragma unroll
        for (int t = 0; t < 8; ++t) {
            const float r = acc[t][v] * scl[v];
            acc[t][v] = r;
            p = fmaf(r, hbf[t], p);
        }
        xyp[v] = hsum16(p);
    }

    float q8[8];
    #pragma unroll
    for (int v = 0; v < 8; ++v) {
        const float xy  = xyp[v];
        const float x2  = rnrm[v] * rnrm[v];
        const float ca  = 1.f + 2.f * xy + y2;
        const float cb  = 1.f - x2;
        const float inv = 1.f / fmaxf(1.f + 2.f * xy + x2 * y2, MINN);
        float p = 0.f;
        #pragma unroll
        for (int t = 0; t < 8; ++t) {
            const float o = (ca * acc[t][v] + cb * hbf[t]) * inv;
            acc[t][v] = o;
            p = fmaf(o, o, p);
        }
        q8[v] = hsum16(p);
    }

    // ---- proj -> logmap0 -> relu ----
    float u8[8];
    #pragma unroll
    for (int v = 0; v < 8; ++v) {
        const float n3 = sqrtf(q8[v]);
        const float pf = (n3 > MAXN) ? (MAXN / n3) : 1.0f;
        const float pn = fmaxf(n3 * pf, MINN);
        const float sl = (artanh_f(pn) / pn) * pf;
        float p = 0.f;
        #pragma unroll
        for (int t = 0; t < 8; ++t) {
            const float u = fmaxf(acc[t][v] * sl, 0.f);
            acc[t][v] = u;
            p = fmaf(u, u, p);
        }
        u8[v] = hsum16(p);
    }

    // ---- expmap0 + final proj, store ----
    #pragma unroll
    for (int v = 0; v < 8; ++v) {
        const float un = fmaxf(sqrtf(u8[v]), MINN);
        const float th = fast_tanh(un);
        const float pf = (th > MAXN) ? (MAXN / th) : 1.0f;
        const float fs = (th / un) * pf;
        const long  m  = row0 + v + 8 * kp;
        #pragma unroll
        for (int t = 0; t < 8; ++t) {
            OUT[m * DOUT + t * 16 + n15] = acc[t][v] * fs;
        }
    }
}

extern "C" void kernel_launch(void* const* d_in, const int* in_sizes, int n_in,
                              void* d_out, int out_size, void* d_ws, size_t ws_size,
                              hipStream_t stream) {
    const float* X = (const float*)d_in[0];
    const float* W = (const float*)d_in[1];
    const float* B = (const float*)d_in[2];
    float* OUT = (float*)d_out;
    const int nrows   = in_sizes[0] / DIN;        // 524288
    const int nblocks = nrows / 128;              // 4096 (16 rows/wave * 8 waves)
    const size_t shmem = (size_t)(DOUT * WSTRIDE + DOUT + 4) * sizeof(float); // ~68 KB (WGP has 320 KB)
    hyp_linear_kernel<<<dim3(nblocks), dim3(256), shmem, stream>>>(X, W, B, OUT, nrows);
}